// LoLAState_38173669327480
// MI455X (gfx1250) — compile-verified
//
#include <hip/hip_runtime.h>

#define Bc 2
#define Cc 512
#define Gc 2048
#define Hc 32
#define Dc 128
#define Fc 64
#define NCAT (Gc + Cc)   // 2560
#define NPAD 4096

// LDS row strides (floats) chosen so the two lane-halves (K=0..1 vs K=2..3,
// row delta = 2) hit bank groups offset by 32 banks: 2*stride % 64 == 32.
#define FK_STRIDE 80     // 64 data + pad
#define V_STRIDE  144    // 128 data + pad

typedef float v2f __attribute__((ext_vector_type(2)));
typedef float v8f __attribute__((ext_vector_type(8)));
typedef int   i4  __attribute__((ext_vector_type(4)));

// gfx1250 async global->LDS path (ASYNCcnt), if this toolchain exposes it.
#if defined(__has_builtin)
#if __has_builtin(__builtin_amdgcn_global_load_async_to_lds_b128) && \
    __has_builtin(__builtin_amdgcn_s_wait_asynccnt)
#define HAVE_ASYNC_LDS 1
#endif
#endif
#ifndef HAVE_ASYNC_LDS
#define HAVE_ASYNC_LDS 0
#endif

#define AS_GLOBAL __attribute__((address_space(1)))
#define AS_LDS    __attribute__((address_space(3)))

// -------------------------------------------------------------------------
// Kernel 1: per-(b,h) bitonic sort of 2560 scores (padded to 4096) in LDS.
// Emits heap_new (sorted top-2048 scores) and top/bot index lists.
// -------------------------------------------------------------------------
__global__ __launch_bounds__(256) void topk_sort_kernel(
    const float* __restrict__ heap_score,   // (B,G,H)
    const float* __restrict__ score_c,      // (B,C,H)
    float* __restrict__ heap_new,           // (B,G,H)
    int* __restrict__ top_idx,              // (B*H, G)
    int* __restrict__ bot_idx)              // (B*H, C)
{
    __shared__ float s_key[NPAD];
    __shared__ int   s_idx[NPAD];
    const int bh  = blockIdx.x;             // 0..63
    const int b   = bh / Hc;
    const int h   = bh % Hc;
    const int tid = threadIdx.x;

    for (int t = tid; t < NPAD; t += 256) {
        float k;
        if (t < Gc)        k = heap_score[(size_t)(b * Gc + t) * Hc + h];
        else if (t < NCAT) k = score_c[(size_t)(b * Cc + (t - Gc)) * Hc + h];
        else               k = -__builtin_inff();
        s_key[t] = k;
        s_idx[t] = t;
    }
    __syncthreads();

    // Bitonic sort, overall descending.
    for (int k = 2; k <= NPAD; k <<= 1) {
        for (int j = k >> 1; j > 0; j >>= 1) {
            for (int i = tid; i < NPAD; i += 256) {
                const int ixj = i ^ j;
                if (ixj > i) {
                    const bool dirAsc = ((i & k) != 0);   // flip => descending
                    const float a = s_key[i];
                    const float c = s_key[ixj];
                    if ((a > c) == dirAsc) {
                        s_key[i] = c; s_key[ixj] = a;
                        const int ti = s_idx[i];
                        s_idx[i] = s_idx[ixj]; s_idx[ixj] = ti;
                    }
                }
            }
            __syncthreads();
        }
    }

    for (int t = tid; t < Gc; t += 256) {
        heap_new[(size_t)(b * Gc + t) * Hc + h] = s_key[t];
        top_idx[(size_t)bh * Gc + t] = s_idx[t];
    }
    for (int t = Gc + tid; t < NCAT; t += 256) {
        bot_idx[(size_t)bh * Cc + (t - Gc)] = s_idx[t];
    }
}

// -------------------------------------------------------------------------
// Kernel 2: gather K/V/FK rows for the top-2048 indices. One wave per row,
// float4 (B128) loads/stores — this is the byte-dominant streaming kernel.
// -------------------------------------------------------------------------
__global__ __launch_bounds__(256) void gather_kernel(
    const float* __restrict__ K_top, const float* __restrict__ V_top,
    const float* __restrict__ FK_top,
    const float* __restrict__ k_c, const float* __restrict__ v_c,
    const float* __restrict__ fk_c,
    const int* __restrict__ top_idx,
    float* __restrict__ outK, float* __restrict__ outV,
    float* __restrict__ outFK)
{
    const int wave = threadIdx.x >> 5;
    const int lane = threadIdx.x & 31;
    const int row  = blockIdx.x * 8 + wave;   // 0 .. B*H*G-1
    const int bh   = row >> 11;               // /G
    const int g    = row & (Gc - 1);
    const int b    = bh / Hc;
    const int h    = bh % Hc;
    const int i    = top_idx[(size_t)bh * Gc + g];

    const float *srcK, *srcV, *srcFK;
    if (i < Gc) {
        const size_t base = (size_t)(b * Gc + i) * Hc + h;
        srcK = K_top + base * Dc; srcV = V_top + base * Dc; srcFK = FK_top + base * Fc;
    } else {
        const size_t base = (size_t)(b * Cc + (i - Gc)) * Hc + h;
        srcK = k_c + base * Dc; srcV = v_c + base * Dc; srcFK = fk_c + base * Fc;
    }
    const size_t dbase = (size_t)(b * Gc + g) * Hc + h;

    const float4 kv = ((const float4*)srcK)[lane];     // 128 f32 = 32 x float4
    const float4 vv = ((const float4*)srcV)[lane];
    ((float4*)(outK + dbase * Dc))[lane] = kv;
    ((float4*)(outV + dbase * Dc))[lane] = vv;
    if (lane < 16) {                                   // 64 f32 = 16 x float4
        const float4 fv = ((const float4*)srcFK)[lane];
        ((float4*)(outFK + dbase * Fc))[lane] = fv;
    }
}

// -------------------------------------------------------------------------
// Kernel 3: H_new = H_sum + FK_bot^T * V_bot  (F=64 x D=128, K=C=512) via
// V_WMMA_F32_16X16X4_F32; S_new = S_sum + colsum(FK_bot). One block per
// (b,h); 8 waves; each wave owns one 16-wide D tile, 4 F tiles stacked.
// LDS staging uses the gfx1250 async global->LDS path when available.
// -------------------------------------------------------------------------
__global__ __launch_bounds__(256) void hsum_wmma_kernel(
    const float* __restrict__ V_top, const float* __restrict__ FK_top,
    const float* __restrict__ v_c,   const float* __restrict__ fk_c,
    const float* __restrict__ H_sum, const float* __restrict__ S_sum,
    const int* __restrict__ bot_idx,
    float* __restrict__ outH, float* __restrict__ outS)
{
    __shared__ float sFK[64 * FK_STRIDE];   // 64 rows x 64 (+pad)  = 20 KB
    __shared__ float sV [64 * V_STRIDE];    // 64 rows x 128 (+pad) = 36 KB
    const int bh   = blockIdx.x;            // 0..63
    const int b    = bh / Hc;
    const int h    = bh % Hc;
    const int tid  = threadIdx.x;
    const int wave = tid >> 5;
    const int lane = tid & 31;
    const int d0   = wave * 16;             // this wave's D tile

    // C/D 16x16 f32 layout: lane<16 -> N=lane, M=r; lane>=16 -> N=lane-16, M=r+8
    const int n    = lane & 15;
    const int moff = (lane >> 4) * 8;

    v8f acc[4];
    {
        const float* Hb = H_sum + (size_t)bh * Fc * Dc;
        #pragma unroll
        for (int ft = 0; ft < 4; ++ft)
            #pragma unroll
            for (int r = 0; r < 8; ++r)
                acc[ft][r] = Hb[(size_t)(ft * 16 + moff + r) * Dc + d0 + n];
    }

    float sacc = 0.0f;

    // A/B 16x4 / 4x16 f32 layout: lanes 0..15 hold K=0 (v0), K=1 (v1);
    // lanes 16..31 hold K=2 (v0), K=3 (v1).
    const int klo = (lane >> 4) * 2;

    for (int chunk = 0; chunk < Cc; chunk += 64) {
        // Stage 64 gathered bottom rows into LDS (one row per wave pass).
        for (int r = wave; r < 64; r += 8) {
            const int i = bot_idx[(size_t)bh * Cc + chunk + r];
            const float *srcV, *srcFK;
            if (i < Gc) {
                const size_t base = (size_t)(b * Gc + i) * Hc + h;
                srcV = V_top + base * Dc; srcFK = FK_top + base * Fc;
            } else {
                const size_t base = (size_t)(b * Cc + (i - Gc)) * Hc + h;
                srcV = v_c + base * Dc; srcFK = fk_c + base * Fc;
            }
#if HAVE_ASYNC_LDS
            // GLOBAL_LOAD_ASYNC_TO_LDS_B128: per-lane 16B global->LDS,
            // tracked by ASYNCcnt (no VGPR round-trip).
            __builtin_amdgcn_global_load_async_to_lds_b128(
                (AS_GLOBAL i4*)(srcV + lane * 4),
                (AS_LDS i4*)(sV + r * V_STRIDE + lane * 4), 0, 0);
            if (lane < 16) {
                __builtin_amdgcn_global_load_async_to_lds_b128(
                    (AS_GLOBAL i4*)(srcFK + lane * 4),
                    (AS_LDS i4*)(sFK + r * FK_STRIDE + lane * 4), 0, 0);
            }
#else
            const float4 vv = ((const float4*)srcV)[lane];
            ((float4*)(sV + r * V_STRIDE))[lane] = vv;
            if (lane < 16) {
                const float4 fv = ((const float4*)srcFK)[lane];
                ((float4*)(sFK + r * FK_STRIDE))[lane] = fv;
            }
#endif
        }
#if HAVE_ASYNC_LDS
        __builtin_amdgcn_s_wait_asynccnt(0);   // our LDS writes have landed
#endif
        __syncthreads();

        // S_new partial: threads 0..63 own one f each (stride-1, no conflicts)
        if (tid < Fc) {
            #pragma unroll 8
            for (int r = 0; r < 64; ++r) sacc += sFK[r * FK_STRIDE + tid];
        }

        // GEMM over this 64-deep chunk, 4 c's per WMMA. EXEC is full here.
        for (int kb = 0; kb < 64; kb += 4) {
            v2f bfrag;
            bfrag.x = sV[(kb + klo    ) * V_STRIDE + d0 + n];
            bfrag.y = sV[(kb + klo + 1) * V_STRIDE + d0 + n];
            #pragma unroll
            for (int ft = 0; ft < 4; ++ft) {
                v2f afrag;
                afrag.x = sFK[(kb + klo    ) * FK_STRIDE + ft * 16 + n];
                afrag.y = sFK[(kb + klo + 1) * FK_STRIDE + ft * 16 + n];
                acc[ft] = __builtin_amdgcn_wmma_f32_16x16x4_f32(
                    false, afrag, false, bfrag, (short)0, acc[ft],
                    false, false);
            }
        }
        __syncthreads();
    }

    {
        float* Ho = outH + (size_t)bh * Fc * Dc;
        #pragma unroll
        for (int ft = 0; ft < 4; ++ft)
            #pragma unroll
            for (int r = 0; r < 8; ++r)
                Ho[(size_t)(ft * 16 + moff + r) * Dc + d0 + n] = acc[ft][r];
    }
    if (tid < Fc)
        outS[(size_t)bh * Fc + tid] = S_sum[(size_t)bh * Fc + tid] + sacc;
}

// -------------------------------------------------------------------------
extern "C" void kernel_launch(void* const* d_in, const int* in_sizes, int n_in,
                              void* d_out, int out_size, void* d_ws, size_t ws_size,
                              hipStream_t stream) {
    (void)in_sizes; (void)n_in; (void)out_size; (void)ws_size;

    const float* heap_score = (const float*)d_in[0];  // (B,G,H)
    const float* K_top      = (const float*)d_in[1];  // (B,G,H,D)
    const float* V_top      = (const float*)d_in[2];  // (B,G,H,D)
    const float* FK_top     = (const float*)d_in[3];  // (B,G,H,F)
    const float* H_sum      = (const float*)d_in[4];  // (B,H,F,D)
    const float* S_sum      = (const float*)d_in[5];  // (B,H,F)
    const float* k_c        = (const float*)d_in[6];  // (B,C,H,D)
    const float* v_c        = (const float*)d_in[7];  // (B,C,H,D)
    const float* fk_c       = (const float*)d_in[8];  // (B,C,H,F)
    const float* score_c    = (const float*)d_in[9];  // (B,C,H)
    // d_in[10] = G (compile-time constant here)

    float* out     = (float*)d_out;
    float* outHeap = out;                               // B*G*H      = 131072
    float* outK    = outHeap + (size_t)Bc * Gc * Hc;    // B*G*H*D    = 16777216
    float* outV    = outK + (size_t)Bc * Gc * Hc * Dc;
    float* outFK   = outV + (size_t)Bc * Gc * Hc * Dc;  // B*G*H*F    = 8388608
    float* outH    = outFK + (size_t)Bc * Gc * Hc * Fc; // B*H*F*D    = 524288
    float* outS    = outH + (size_t)Bc * Hc * Fc * Dc;  // B*H*F      = 4096

    int* top_idx = (int*)d_ws;                          // B*H*G ints = 512 KB
    int* bot_idx = top_idx + (size_t)Bc * Hc * Gc;      // B*H*C ints = 128 KB

    topk_sort_kernel<<<Bc * Hc, 256, 0, stream>>>(
        heap_score, score_c, outHeap, top_idx, bot_idx);

    gather_kernel<<<(Bc * Hc * Gc) / 8, 256, 0, stream>>>(
        K_top, V_top, FK_top, k_c, v_c, fk_c, top_idx, outK, outV, outFK);

    hsum_wmma_kernel<<<Bc * Hc, 256, 0, stream>>>(
        V_top, FK_top, v_c, fk_c, H_sum, S_sum, bot_idx, outH, outS);
}